// EncoderLayer_68917045232302
// MI455X (gfx1250) — compile-verified
//
#include <hip/hip_runtime.h>
#include <math.h>

typedef __attribute__((ext_vector_type(2))) float v2f;
typedef __attribute__((ext_vector_type(8))) float v8f;

#define B_   4
#define S_   256
#define H_   256
#define NH_  8
#define D_   32
#define PF_  1024
#define INV_SCALE_ 0.17677669529663687f   // 1/sqrt(32)

// ---------------------------------------------------------------------------
// fp32 WMMA: D(16x16) = A(16x4) * B(4x16) + C   (V_WMMA_F32_16X16X4_F32)
// A frag: lane l holds A[l&15][2*(l>>4)+v] in element v
// B frag: lane l holds B[2*(l>>4)+v][l&15] in element v
// C/D:    lane l, elem r holds D[r + 8*(l>>4)][l&15]
// ---------------------------------------------------------------------------
__device__ __forceinline__ v8f wmma4(v2f a, v2f b, v8f c) {
  return __builtin_amdgcn_wmma_f32_16x16x4_f32(
      false, a, false, b, (short)0, c, false, false);
}

// ---------------------------------------------------------------------------
// CDNA5 async copy: per-lane 16B global -> LDS, tracked by ASYNCcnt.
// LDS flat pointers keep the byte offset in addr[31:0] (aperture in [63:32]),
// so truncation yields the LDS address the instruction wants.
// ---------------------------------------------------------------------------
__device__ __forceinline__ void async_b128(void* lds_ptr, const float* g) {
  unsigned loff = (unsigned)(unsigned long long)lds_ptr;
  asm volatile("global_load_async_to_lds_b128 %0, %1, off"
               :: "v"(loff), "v"(g) : "memory");
}
__device__ __forceinline__ void wait_async0() {
  asm volatile("s_wait_asynccnt 0x0" ::: "memory");
}

// ---------------------------------------------------------------------------
// Fold W1/W2 (per-head 32x32) into the K/Q projection weights:
//   WkW1[:, h*32+j] = Wk[:, h*32:h*32+32] @ W1[:, j];  bk1 = bk@W1 + b1 (tiled)
// ---------------------------------------------------------------------------
__global__ void fold_weights_kernel(const float* __restrict__ Wk, const float* __restrict__ bk,
                                    const float* __restrict__ Wq, const float* __restrict__ bq,
                                    const float* __restrict__ W1, const float* __restrict__ b1,
                                    const float* __restrict__ W2, const float* __restrict__ b2,
                                    float* __restrict__ WkW1, float* __restrict__ bk1,
                                    float* __restrict__ WqW2, float* __restrict__ bq2) {
  int gid   = blockIdx.x * blockDim.x + threadIdx.x;  // [0, 2*H*H)
  int which = gid >> 16;
  int rem   = gid & 0xFFFF;
  int i     = rem >> 8;
  int col   = rem & 255;
  int h     = col >> 5, j = col & 31;
  const float* Ws = which ? Wq : Wk;
  const float* Wm = which ? W2 : W1;
  float acc = 0.f;
  #pragma unroll
  for (int d = 0; d < 32; ++d)
    acc += Ws[i * H_ + h * 32 + d] * Wm[d * 32 + j];
  (which ? WqW2 : WkW1)[i * H_ + col] = acc;
  if (i == 0) {
    const float* bs = which ? bq : bk;
    const float* bm = which ? b2 : b1;
    float bacc = bm[j];
    #pragma unroll
    for (int d = 0; d < 32; ++d)
      bacc += bs[h * 32 + d] * Wm[d * 32 + j];
    (which ? bq2 : bk1)[col] = bacc;
  }
}

// ---------------------------------------------------------------------------
// C[M,N] = act(A[M,K] @ W[K,N] + bias[N]).  Block tile 128x64, 8 waves
// (wave w -> rows w*16..w*16+15, all 64 cols).  K chunked by 32, A/W chunks
// staged to LDS with async b128 copies, double-buffered.
// grid = (N/64, M/128), blockDim = 256.
// ---------------------------------------------------------------------------
template <int K, int N>
__global__ __launch_bounds__(256) void gemm_bias_kernel(
    const float* __restrict__ A, const float* __restrict__ W,
    const float* __restrict__ bias, float* __restrict__ C, int relu) {
  constexpr int KC  = 32;
  constexpr int NCH = K / KC;
  __shared__ float sA[2][128][36];  // 36*4B row stride: multiple of 16B, conflict-free frag reads
  __shared__ float sW[2][32][72];   // 2 rows apart -> +16 banks: halves disjoint

  const int t    = threadIdx.x;
  const int lane = t & 31;
  const int wave = t >> 5;
  const int lm   = lane & 15;
  const int hi   = lane >> 4;
  const int m0   = blockIdx.y * 128;
  const int n0   = blockIdx.x * 64;

  auto stage = [&](int c, int buf) {
    const int kb = c * KC;
    #pragma unroll
    for (int p = 0; p < 4; ++p) {            // A chunk: 128x32 = 1024 xfers
      const int e = p * 256 + t;
      const int r = e >> 3, c4 = (e & 7) * 4;
      async_b128(&sA[buf][r][c4], A + (size_t)(m0 + r) * K + kb + c4);
    }
    #pragma unroll
    for (int p = 0; p < 2; ++p) {            // W chunk: 32x64 = 512 xfers
      const int e = p * 256 + t;
      const int r = e >> 4, c4 = (e & 15) * 4;
      async_b128(&sW[buf][r][c4], W + (size_t)(kb + r) * N + n0 + c4);
    }
  };

  v8f acc[4] = {};
  stage(0, 0);
  wait_async0();
  __syncthreads();

  for (int c = 0; c < NCH; ++c) {
    const int cur = c & 1;
    if (c + 1 < NCH) stage(c + 1, cur ^ 1);  // async prefetch overlaps compute
    #pragma unroll
    for (int kk = 0; kk < KC; kk += 4) {
      const int k2 = kk + 2 * hi;
      v2f a; a.x = sA[cur][wave * 16 + lm][k2];
             a.y = sA[cur][wave * 16 + lm][k2 + 1];
      #pragma unroll
      for (int tt = 0; tt < 4; ++tt) {
        v2f b; b.x = sW[cur][k2][tt * 16 + lm];
               b.y = sW[cur][k2 + 1][tt * 16 + lm];
        acc[tt] = wmma4(a, b, acc[tt]);
      }
    }
    wait_async0();
    __syncthreads();
  }

  #pragma unroll
  for (int tt = 0; tt < 4; ++tt) {
    const int col = n0 + tt * 16 + lm;
    const float bv = bias[col];
    #pragma unroll
    for (int r = 0; r < 8; ++r) {
      float v = acc[tt][r] + bv;
      if (relu) v = fmaxf(v, 0.f);
      C[(size_t)(m0 + r + 8 * hi) * N + col] = v;
    }
  }
}

// ---------------------------------------------------------------------------
// Out = LayerNorm(A @ W + bias + R), N == 256.  Block tile 16x256 (4 waves,
// wave w -> cols w*64..w*64+63), async LDS staging (single-buffered), fused
// residual + LayerNorm over the full row strip.  grid = M/16, blockDim = 128.
// ---------------------------------------------------------------------------
template <int K>
__global__ __launch_bounds__(128) void gemm_res_ln_kernel(
    const float* __restrict__ A, const float* __restrict__ W,
    const float* __restrict__ bias, const float* __restrict__ R,
    const float* __restrict__ g, const float* __restrict__ be,
    float* __restrict__ Out) {
  constexpr int KC  = 32;
  constexpr int NCH = K / KC;
  __shared__ float sA[16][36];
  __shared__ float sW[32][264];    // 264: 16B-aligned rows, +2 rows -> +16 banks
  __shared__ float srow[16][257];

  const int t    = threadIdx.x;
  const int lane = t & 31;
  const int wave = t >> 5;
  const int lm   = lane & 15;
  const int hi   = lane >> 4;
  const int m0   = blockIdx.x * 16;
  const int n0   = wave * 64;

  v8f acc[4] = {};
  for (int c = 0; c < NCH; ++c) {
    const int kb = c * KC;
    if (c) __syncthreads();                  // protect single buffer
    {                                        // A chunk: 16x32 = 128 xfers
      const int r = t >> 3, c4 = (t & 7) * 4;
      async_b128(&sA[r][c4], A + (size_t)(m0 + r) * K + kb + c4);
    }
    #pragma unroll
    for (int p = 0; p < 16; ++p) {           // W chunk: 32x256 = 2048 xfers
      const int e = p * 128 + t;
      const int r = e >> 6, c4 = (e & 63) * 4;
      async_b128(&sW[r][c4], W + (size_t)(kb + r) * 256 + c4);
    }
    wait_async0();
    __syncthreads();

    #pragma unroll
    for (int kk = 0; kk < KC; kk += 4) {
      const int k2 = kk + 2 * hi;
      v2f a; a.x = sA[lm][k2]; a.y = sA[lm][k2 + 1];
      #pragma unroll
      for (int tt = 0; tt < 4; ++tt) {
        v2f b; b.x = sW[k2][n0 + tt * 16 + lm];
               b.y = sW[k2 + 1][n0 + tt * 16 + lm];
        acc[tt] = wmma4(a, b, acc[tt]);
      }
    }
  }

  #pragma unroll
  for (int tt = 0; tt < 4; ++tt) {
    const int col = n0 + tt * 16 + lm;
    const float bv = bias[col];
    #pragma unroll
    for (int r = 0; r < 8; ++r) {
      const int row = r + 8 * hi;
      srow[row][col] = acc[tt][r] + bv + R[(size_t)(m0 + row) * 256 + col];
    }
  }
  __syncthreads();

  // LayerNorm: 8 threads per row, shuffle-reduce within 8-lane groups
  const int r = t >> 3;
  const int j = t & 7;
  float s1 = 0.f, s2 = 0.f;
  for (int i = 0; i < 32; ++i) {
    const float x = srow[r][j + 8 * i];
    s1 += x; s2 += x * x;
  }
  #pragma unroll
  for (int mk = 4; mk >= 1; mk >>= 1) {
    s1 += __shfl_xor(s1, mk, 8);
    s2 += __shfl_xor(s2, mk, 8);
  }
  const float mean = s1 * (1.0f / 256.0f);
  const float var  = s2 * (1.0f / 256.0f) - mean * mean;
  const float rstd = rsqrtf(var + 1e-5f);
  for (int i = 0; i < 32; ++i) {
    const int col = j + 8 * i;
    const float x = (srow[r][col] - mean) * rstd;
    Out[(size_t)(m0 + r) * 256 + col] = x * g[col] + be[col];
  }
}

// ---------------------------------------------------------------------------
// Additive attention, per (b, h, 16-q tile):
//   energy[q,k] = (sum_d vw[d]*tanh(W1K[k,d]+W2Q[q,d]) + vb)/sqrt(D), mask,
//   softmax over k, then X[q,:] = attn[q,:] @ V via WMMA.
// W1K/W2Q head slices staged to LDS with async b128.  blockDim = 256.
// ---------------------------------------------------------------------------
__global__ __launch_bounds__(256) void attention_kernel(
    const float* __restrict__ W1K, const float* __restrict__ W2Q,
    const float* __restrict__ V,   const float* __restrict__ vw,
    const float* __restrict__ vb_ptr, const int* __restrict__ mask,
    float* __restrict__ X) {
  __shared__ float s_w1k[256][36];  // stride 36: 16B-aligned + conflict-free reads
  __shared__ float s_att[16][257];
  __shared__ float s_w2q[16][36];
  __shared__ float s_vw[32];

  const int b  = blockIdx.z, h = blockIdx.y;
  const int q0 = blockIdx.x * 16;
  const int t  = threadIdx.x;

  #pragma unroll
  for (int p = 0; p < 8; ++p) {              // W1K slice: 256x32 = 2048 xfers
    const int e = p * 256 + t;
    const int k = e >> 3, c4 = (e & 7) * 4;
    async_b128(&s_w1k[k][c4], W1K + (size_t)(b * 256 + k) * H_ + h * 32 + c4);
  }
  if (t < 128) {                             // W2Q tile: 16x32 = 128 xfers
    const int q = t >> 3, c4 = (t & 7) * 4;
    async_b128(&s_w2q[q][c4], W2Q + (size_t)(b * 256 + q0 + q) * H_ + h * 32 + c4);
  }
  if (t < 32) s_vw[t] = vw[t];
  wait_async0();
  __syncthreads();

  const float vb = vb_ptr[0];

  // ---- energy: 16 threads per q row, each covers k = j + 16*ki ----
  {
    const int q = t >> 4, j = t & 15;
    for (int ki = 0; ki < 16; ++ki) {
      const int k = 16 * ki + j;
      float acc = 0.f;
      #pragma unroll 8
      for (int d = 0; d < 32; ++d)
        acc += s_vw[d] * tanhf(s_w1k[k][d] + s_w2q[q][d]);
      float e = (acc + vb) * INV_SCALE_;
      if (mask[b * S_ + k] == 0) e = -1e10f;
      s_att[q][k] = e;
    }
  }
  __syncthreads();

  // ---- softmax per row: shuffle-reduce within the 16 lanes of each row ----
  {
    const int q = t >> 4, j = t & 15;
    float m = -3.4e38f;
    for (int ki = 0; ki < 16; ++ki) m = fmaxf(m, s_att[q][16 * ki + j]);
    #pragma unroll
    for (int mk = 8; mk >= 1; mk >>= 1) m = fmaxf(m, __shfl_xor(m, mk, 16));
    float s = 0.f;
    for (int ki = 0; ki < 16; ++ki) {
      const int k = 16 * ki + j;
      const float e = __expf(s_att[q][k] - m);
      s_att[q][k] = e;
      s += e;
    }
    #pragma unroll
    for (int mk = 8; mk >= 1; mk >>= 1) s += __shfl_xor(s, mk, 16);
    const float inv = 1.0f / s;
    for (int ki = 0; ki < 16; ++ki) s_att[q][16 * ki + j] *= inv;
  }
  __syncthreads();

  // ---- attn(16x256) @ V(256x32): waves 0,1 take the two 16-col tiles
  // (wave-uniform branch -> EXEC all-1s inside, as WMMA requires) ----
  const int wave = t >> 5;
  if (wave < 2) {
    const int lane = t & 31, lm = lane & 15, hi = lane >> 4;
    const int n0 = wave * 16;
    v8f acc = {};
    for (int kb = 0; kb < 256; kb += 4) {
      const int kk = kb + 2 * hi;
      v2f a;  a.x = s_att[lm][kk];  a.y = s_att[lm][kk + 1];
      v2f bb;
      bb.x = V[(size_t)(b * 256 + kk)     * H_ + h * 32 + n0 + lm];
      bb.y = V[(size_t)(b * 256 + kk + 1) * H_ + h * 32 + n0 + lm];
      acc = wmma4(a, bb, acc);
    }
    #pragma unroll
    for (int r = 0; r < 8; ++r) {
      const int row = r + 8 * hi;
      X[(size_t)(b * 256 + q0 + row) * H_ + h * 32 + n0 + lm] = acc[r];
    }
  }
}

// ---------------------------------------------------------------------------
extern "C" void kernel_launch(void* const* d_in, const int* in_sizes, int n_in,
                              void* d_out, int out_size, void* d_ws, size_t ws_size,
                              hipStream_t stream) {
  const float* src  = (const float*)d_in[0];
  const int*   mask = (const int*)  d_in[1];
  const float* Wq = (const float*)d_in[2];   const float* bq = (const float*)d_in[3];
  const float* Wk = (const float*)d_in[4];   const float* bk = (const float*)d_in[5];
  const float* Wv = (const float*)d_in[6];   const float* bv = (const float*)d_in[7];
  const float* Wo = (const float*)d_in[8];   const float* bo = (const float*)d_in[9];
  const float* W1 = (const float*)d_in[10];  const float* b1 = (const float*)d_in[11];
  const float* W2 = (const float*)d_in[12];  const float* b2 = (const float*)d_in[13];
  const float* vw = (const float*)d_in[14];  const float* vb = (const float*)d_in[15];
  const float* g1 = (const float*)d_in[16];  const float* be1 = (const float*)d_in[17];
  const float* g2 = (const float*)d_in[18];  const float* be2 = (const float*)d_in[19];
  const float* Wf1 = (const float*)d_in[20]; const float* bf1 = (const float*)d_in[21];
  const float* Wf2 = (const float*)d_in[22]; const float* bf2 = (const float*)d_in[23];

  float* ws   = (float*)d_ws;
  float* WkW1 = ws;  ws += H_ * H_;
  float* WqW2 = ws;  ws += H_ * H_;
  float* bk1  = ws;  ws += H_;
  float* bq2  = ws;  ws += H_;
  float* W1K  = ws;  ws += B_ * S_ * H_;
  float* W2Q  = ws;  ws += B_ * S_ * H_;
  float* Vbuf = ws;  ws += B_ * S_ * H_;
  float* Xbuf = ws;  ws += B_ * S_ * H_;
  float* Hbuf = ws;  ws += B_ * S_ * H_;
  float* FF1  = ws;  ws += B_ * S_ * PF_;
  float* out  = (float*)d_out;

  const int M = B_ * S_;  // 1024

  // 1) fold per-head W1/W2 into the K/Q projection weights
  fold_weights_kernel<<<512, 256, 0, stream>>>(Wk, bk, Wq, bq, W1, b1, W2, b2,
                                               WkW1, bk1, WqW2, bq2);
  // 2) fused projections: W1K, W2Q, V   (each [B*S, H])
  gemm_bias_kernel<256, 256><<<dim3(4, 8), 256, 0, stream>>>(src, WkW1, bk1, W1K, 0);
  gemm_bias_kernel<256, 256><<<dim3(4, 8), 256, 0, stream>>>(src, WqW2, bq2, W2Q, 0);
  gemm_bias_kernel<256, 256><<<dim3(4, 8), 256, 0, stream>>>(src, Wv,   bv,  Vbuf, 0);
  // 3) additive attention + softmax + attn@V (WMMA)
  attention_kernel<<<dim3(S_ / 16, NH_, B_), 256, 0, stream>>>(W1K, W2Q, Vbuf, vw, vb, mask, Xbuf);
  // 4) Wo projection + residual + LayerNorm1
  gemm_res_ln_kernel<256><<<M / 16, 128, 0, stream>>>(Xbuf, Wo, bo, src, g1, be1, Hbuf);
  // 5) FFN1 (ReLU)
  gemm_bias_kernel<256, 1024><<<dim3(16, 8), 256, 0, stream>>>(Hbuf, Wf1, bf1, FF1, 1);
  // 6) FFN2 + residual + LayerNorm2 -> output
  gemm_res_ln_kernel<1024><<<M / 16, 128, 0, stream>>>(FF1, Wf2, bf2, Hbuf, g2, be2, out);

  (void)in_sizes; (void)n_in; (void)out_size; (void)ws_size;
}